// vlad_encoder_43533788512263
// MI455X (gfx1250) — compile-verified
//
#include <hip/hip_runtime.h>
#include <math.h>

// ---------------------------------------------------------------------------
// NetVLAD encoder for MI455X (gfx1250, wave32, WMMA).
// Shapes: B=16, N=1024, Din=1024, H=512, K=64, Dout=1024.
// GEMMs use V_WMMA_F32_16X16X4_F32 (full fp32) with 2x2 / 1x4 register
// blocking so the VMEM-instruction rate stays below the WMMA issue rate.
// fc2_w (134MB, single-use) is streamed with non-temporal loads.
// ---------------------------------------------------------------------------

#define NV_EPS 1e-12f

typedef __attribute__((ext_vector_type(2))) float v2f;
typedef __attribute__((ext_vector_type(8))) float v8f;

static __device__ __forceinline__ v8f wmma4(v2f a, v2f b, v8f c) {
  return __builtin_amdgcn_wmma_f32_16x16x4_f32(false, a, false, b, (short)0, c,
                                               false, false);
}

static __device__ __forceinline__ float warp_sum32(float v) {
  v += __shfl_xor(v, 16);
  v += __shfl_xor(v, 8);
  v += __shfl_xor(v, 4);
  v += __shfl_xor(v, 2);
  v += __shfl_xor(v, 1);
  return v;
}

// ---------------------------------------------------------------------------
// Kernel 1: h = relu(input @ fc1_w^T + fc1_b)   [16384,1024]x[512,1024]^T
// One wave = one 32x32 output block (2x2 WMMA tiles, fragments reused).
// 512 Mtiles * 16 Ntiles = 8192 wave jobs.
// ---------------------------------------------------------------------------
__global__ __launch_bounds__(256) void k_fc1(const float* __restrict__ x,
                                             const float* __restrict__ w,
                                             const float* __restrict__ bias,
                                             float* __restrict__ h) {
  const int lane = threadIdx.x & 31;
  const int wave = threadIdx.x >> 5;
  const int tile = blockIdx.x * 8 + wave;   // 0..8191
  const int mt = tile >> 4;                 // 0..511  (32-row block)
  const int nt = tile & 15;                 // 0..15   (32-col block)
  const int half = lane >> 4;               // K sub-block 0/1
  const int l = lane & 15;
  const int row0 = mt * 32;
  const int col0 = nt * 32;

  const float* ap0 = x + (size_t)(row0 + l) * 1024 + half * 2;
  const float* ap1 = ap0 + (size_t)16 * 1024;
  const float* bp0 = w + (size_t)(col0 + l) * 1024 + half * 2;
  const float* bp1 = bp0 + (size_t)16 * 1024;

  v8f c00 = {}, c01 = {}, c10 = {}, c11 = {};
#pragma unroll 4
  for (int k0 = 0; k0 < 1024; k0 += 4) {
    const v2f a0 = *(const v2f*)(ap0 + k0);
    const v2f a1 = *(const v2f*)(ap1 + k0);
    const v2f b0 = *(const v2f*)(bp0 + k0);
    const v2f b1 = *(const v2f*)(bp1 + k0);
    c00 = wmma4(a0, b0, c00);
    c01 = wmma4(a0, b1, c01);
    c10 = wmma4(a1, b0, c10);
    c11 = wmma4(a1, b1, c11);
  }

  const int cA = col0 + l;
  const int cB = col0 + 16 + l;
  const float bcA = bias[cA];
  const float bcB = bias[cB];
#pragma unroll
  for (int r = 0; r < 8; ++r) {
    const int rA = row0 + r + half * 8;
    const int rB = rA + 16;
    float v;
    v = c00[r] + bcA; h[(size_t)rA * 512 + cA] = v > 0.f ? v : 0.f;
    v = c01[r] + bcB; h[(size_t)rA * 512 + cB] = v > 0.f ? v : 0.f;
    v = c10[r] + bcA; h[(size_t)rB * 512 + cA] = v > 0.f ? v : 0.f;
    v = c11[r] + bcB; h[(size_t)rB * 512 + cB] = v > 0.f ? v : 0.f;
  }
}

// ---------------------------------------------------------------------------
// Kernel 2: per-(b,n) L2 normalize over H=512, in place. One wave per row.
// ---------------------------------------------------------------------------
__global__ __launch_bounds__(256) void k_norm_h(float* __restrict__ h) {
  const int lane = threadIdx.x & 31;
  const int wave = threadIdx.x >> 5;
  const int row = blockIdx.x * 8 + wave;    // 0..16383
  float* p = h + (size_t)row * 512;
  float s = 0.f;
#pragma unroll
  for (int i = 0; i < 4; ++i) {
    const float4 v = *(const float4*)(p + i * 128 + lane * 4);
    s += v.x * v.x + v.y * v.y + v.z * v.z + v.w * v.w;
  }
  s = warp_sum32(s);
  const float scale = 1.f / fmaxf(sqrtf(s), NV_EPS);
#pragma unroll
  for (int i = 0; i < 4; ++i) {
    float4 v = *(const float4*)(p + i * 128 + lane * 4);
    v.x *= scale; v.y *= scale; v.z *= scale; v.w *= scale;
    *(float4*)(p + i * 128 + lane * 4) = v;
  }
}

// ---------------------------------------------------------------------------
// Kernel 3: soft-assignment a[b,n,k] = softmax_k(conv_w @ xn + conv_b).
// 64-thread block handles 8 (b,n) rows; each conv_w element is reused
// across 8 rows (cuts conv_w L2 traffic from 2GB to 256MB).
// ---------------------------------------------------------------------------
__global__ __launch_bounds__(64) void k_assign(const float* __restrict__ xn,
                                               const float* __restrict__ cw,
                                               const float* __restrict__ cb,
                                               float* __restrict__ a) {
  __shared__ float xrow[8][512];
  __shared__ float sm[8][64];
  const int k = threadIdx.x;                   // 0..63
  const size_t base = (size_t)blockIdx.x * 8;  // first descriptor row

  // Stage 8 rows of xn (16KB) into LDS: 1024 float4, 16 per thread.
  const float4* src = (const float4*)(xn + base * 512);
  float4* dst = (float4*)&xrow[0][0];
#pragma unroll
  for (int i = 0; i < 16; ++i) dst[k + i * 64] = src[k + i * 64];
  __syncthreads();

  float dot[8] = {0.f, 0.f, 0.f, 0.f, 0.f, 0.f, 0.f, 0.f};
  const float* wr = cw + (size_t)k * 512;
  for (int hh = 0; hh < 512; hh += 4) {
    const float4 wv = *(const float4*)(wr + hh);
#pragma unroll
    for (int r = 0; r < 8; ++r) {
      dot[r] += wv.x * xrow[r][hh] + wv.y * xrow[r][hh + 1] +
                wv.z * xrow[r][hh + 2] + wv.w * xrow[r][hh + 3];
    }
  }
  const float cbk = cb[k];
#pragma unroll
  for (int r = 0; r < 8; ++r) sm[r][k] = dot[r] + cbk;
  __syncthreads();

  float e[8];
#pragma unroll
  for (int r = 0; r < 8; ++r) {
    float mx = -INFINITY;
    for (int i = 0; i < 64; ++i) mx = fmaxf(mx, sm[r][i]);
    e[r] = expf(sm[r][k] - mx);
  }
  __syncthreads();
#pragma unroll
  for (int r = 0; r < 8; ++r) sm[r][k] = e[r];
  __syncthreads();
#pragma unroll
  for (int r = 0; r < 8; ++r) {
    float sum = 0.f;
    for (int i = 0; i < 64; ++i) sum += sm[r][i];
    a[(base + r) * 64 + k] = e[r] / sum;
  }
}

// ---------------------------------------------------------------------------
// Kernel 3b: asum[b,k] = sum_n a[b,n,k]. 16 blocks x 64 threads (coalesced).
// ---------------------------------------------------------------------------
__global__ __launch_bounds__(64) void k_asum(const float* __restrict__ a,
                                             float* __restrict__ asum) {
  const int b = blockIdx.x;
  const int k = threadIdx.x;
  const float* p = a + (size_t)b * 1024 * 64 + k;
  float s = 0.f;
  for (int n = 0; n < 1024; ++n) s += p[n * 64];
  asum[b * 64 + k] = s;
}

// ---------------------------------------------------------------------------
// Kernel 4: vlad[b,k,h] = sum_n a[b,n,k]*xn[b,n,h] - asum[b,k]*cent[k,h].
// Per-b GEMM a^T(64x1024) x xn(1024x512); 2x2 WMMA blocking (32x32/wave).
// 16 b * 2 Mtiles * 16 Ntiles = 512 wave jobs.
// ---------------------------------------------------------------------------
__global__ __launch_bounds__(256) void k_vlad(const float* __restrict__ a,
                                              const float* __restrict__ xn,
                                              const float* __restrict__ asum,
                                              const float* __restrict__ cent,
                                              float* __restrict__ vlad) {
  const int lane = threadIdx.x & 31;
  const int wave = threadIdx.x >> 5;
  const int tile = blockIdx.x * 8 + wave;   // 0..511
  const int b = tile >> 5;                  // 0..15
  const int rem = tile & 31;
  const int mt = rem >> 4;                  // 0..1   (32-cluster block)
  const int nt = rem & 15;                  // 0..15  (32-H block)
  const int half = lane >> 4;
  const int l = lane & 15;

  // A fragment: A[m=k_cluster][kk=n] = a[(b*1024 + n)*64 + cluster]
  const float* ap0 = a + ((size_t)b * 1024 + half * 2) * 64 + mt * 32 + l;
  const float* ap1 = ap0 + 16;
  // B fragment: B[kk=n][nn=h] = xn[(b*1024 + n)*512 + h]
  const float* bp0 = xn + ((size_t)b * 1024 + half * 2) * 512 + nt * 32 + l;
  const float* bp1 = bp0 + 16;

  v8f c00 = {}, c01 = {}, c10 = {}, c11 = {};
#pragma unroll 4
  for (int n0 = 0; n0 < 1024; n0 += 4) {
    const v2f a0 = {ap0[n0 * 64], ap0[n0 * 64 + 64]};
    const v2f a1 = {ap1[n0 * 64], ap1[n0 * 64 + 64]};
    const v2f b0 = {bp0[n0 * 512], bp0[n0 * 512 + 512]};
    const v2f b1 = {bp1[n0 * 512], bp1[n0 * 512 + 512]};
    c00 = wmma4(a0, b0, c00);
    c01 = wmma4(a0, b1, c01);
    c10 = wmma4(a1, b0, c10);
    c11 = wmma4(a1, b1, c11);
  }

  const int h0 = nt * 32 + l;
  const int h1 = h0 + 16;
#pragma unroll
  for (int r = 0; r < 8; ++r) {
    const int k0 = mt * 32 + r + half * 8;
    const int k1 = k0 + 16;
    const float s0 = asum[b * 64 + k0];
    const float s1 = asum[b * 64 + k1];
    float* vb = vlad + (size_t)b * 32768;
    vb[(size_t)k0 * 512 + h0] = c00[r] - s0 * cent[(size_t)k0 * 512 + h0];
    vb[(size_t)k0 * 512 + h1] = c01[r] - s0 * cent[(size_t)k0 * 512 + h1];
    vb[(size_t)k1 * 512 + h0] = c10[r] - s1 * cent[(size_t)k1 * 512 + h0];
    vb[(size_t)k1 * 512 + h1] = c11[r] - s1 * cent[(size_t)k1 * 512 + h1];
  }
}

// ---------------------------------------------------------------------------
// Kernel 5: intra-normalize each (b,k) row of 512 in place; accumulate
// per-batch global sum of squares (post-normalization) into gsum[b].
// ---------------------------------------------------------------------------
__global__ __launch_bounds__(256) void k_vlad_norm(float* __restrict__ vlad,
                                                   float* __restrict__ gsum) {
  const int lane = threadIdx.x & 31;
  const int wave = threadIdx.x >> 5;
  const int row = blockIdx.x * 8 + wave;    // 0..1023 == b*64 + k
  float* p = vlad + (size_t)row * 512;
  float s = 0.f;
#pragma unroll
  for (int i = 0; i < 4; ++i) {
    const float4 v = *(const float4*)(p + i * 128 + lane * 4);
    s += v.x * v.x + v.y * v.y + v.z * v.z + v.w * v.w;
  }
  s = warp_sum32(s);
  const float scale = 1.f / fmaxf(sqrtf(s), NV_EPS);
#pragma unroll
  for (int i = 0; i < 4; ++i) {
    float4 v = *(const float4*)(p + i * 128 + lane * 4);
    v.x *= scale; v.y *= scale; v.z *= scale; v.w *= scale;
    *(float4*)(p + i * 128 + lane * 4) = v;
  }
  if (lane == 0) atomicAdd(&gsum[row >> 6], s * scale * scale);
}

// ---------------------------------------------------------------------------
// Kernel 6: fc2 split-K WMMA GEMM: acc2[b,d] += vladn[b,kc] . fc2_w[d,kc].
// M=16 (batch) = one WMMA tile; 1x4 N-blocking (64 d columns per wave) so
// the A fragment is reused 4x. 16 d-groups x 64 K-chunks of 512 = 1024 jobs.
// fc2_w is 134MB single-use -> non-temporal loads keep it out of L2.
// ---------------------------------------------------------------------------
__global__ __launch_bounds__(256) void k_fc2(const float* __restrict__ vlad,
                                             const float* __restrict__ w,
                                             float* __restrict__ acc2) {
  const int lane = threadIdx.x & 31;
  const int wave = threadIdx.x >> 5;
  const int job = blockIdx.x * 8 + wave;    // 0..1023
  const int dg = job >> 6;                  // 0..15  (64-col d group)
  const int kc = job & 63;                  // 0..63  (512-wide K chunk)
  const int half = lane >> 4;
  const int l = lane & 15;

  const float* ap = vlad + (size_t)l * 32768 + kc * 512 + half * 2;
  const float* bp0 = w + (size_t)(dg * 64 + l) * 32768 + kc * 512 + half * 2;
  const float* bp1 = bp0 + (size_t)16 * 32768;
  const float* bp2 = bp0 + (size_t)32 * 32768;
  const float* bp3 = bp0 + (size_t)48 * 32768;

  v8f c0 = {}, c1 = {}, c2 = {}, c3 = {};
#pragma unroll 4
  for (int k0 = 0; k0 < 512; k0 += 4) {
    const v2f av = *(const v2f*)(ap + k0);
    const v2f b0 = __builtin_nontemporal_load((const v2f*)(bp0 + k0));
    const v2f b1 = __builtin_nontemporal_load((const v2f*)(bp1 + k0));
    const v2f b2 = __builtin_nontemporal_load((const v2f*)(bp2 + k0));
    const v2f b3 = __builtin_nontemporal_load((const v2f*)(bp3 + k0));
    c0 = wmma4(av, b0, c0);
    c1 = wmma4(av, b1, c1);
    c2 = wmma4(av, b2, c2);
    c3 = wmma4(av, b3, c3);
  }

  const int d0 = dg * 64 + l;
#pragma unroll
  for (int r = 0; r < 8; ++r) {
    const int bb = (r + half * 8) * 1024;   // batch row base
    atomicAdd(&acc2[bb + d0], c0[r]);
    atomicAdd(&acc2[bb + d0 + 16], c1[r]);
    atomicAdd(&acc2[bb + d0 + 32], c2[r]);
    atomicAdd(&acc2[bb + d0 + 48], c3[r]);
  }
}

// ---------------------------------------------------------------------------
// Kernel 7: out[b,d] = relu(acc2[b,d] / max(sqrt(gsum[b]),eps) + fc2_b[d])
// ---------------------------------------------------------------------------
__global__ __launch_bounds__(256) void k_final(const float* __restrict__ acc2,
                                               const float* __restrict__ gsum,
                                               const float* __restrict__ bias,
                                               float* __restrict__ out) {
  const int i = blockIdx.x * 256 + threadIdx.x;   // 0..16383
  const int b = i >> 10;
  const int d = i & 1023;
  const float inv = 1.f / fmaxf(sqrtf(gsum[b]), NV_EPS);
  const float v = acc2[i] * inv + bias[d];
  out[i] = v > 0.f ? v : 0.f;
}

// ---------------------------------------------------------------------------
extern "C" void kernel_launch(void* const* d_in, const int* in_sizes, int n_in,
                              void* d_out, int out_size, void* d_ws,
                              size_t ws_size, hipStream_t stream) {
  (void)in_sizes; (void)n_in; (void)out_size; (void)ws_size;
  const float* input = (const float*)d_in[0];  // [16,1024,1024]
  const float* fc1_w = (const float*)d_in[1];  // [512,1024]
  const float* fc1_b = (const float*)d_in[2];  // [512]
  const float* conv_w = (const float*)d_in[3]; // [64,512]
  const float* conv_b = (const float*)d_in[4]; // [64]
  const float* cent = (const float*)d_in[5];   // [64,512]
  const float* fc2_w = (const float*)d_in[6];  // [1024,32768]
  const float* fc2_b = (const float*)d_in[7];  // [1024]
  float* out = (float*)d_out;                  // [16,1024]

  float* ws = (float*)d_ws;
  float* h = ws;                       // 16384*512      = 8388608
  float* a = h + 8388608;              // 16384*64       = 1048576
  float* asum = a + 1048576;           // 16*64          = 1024
  float* vlad = asum + 1024;           // 16*64*512      = 524288
  float* gsum = vlad + 524288;         // 16
  float* acc2 = gsum + 16;             // 16*1024        = 16384

  // Zero the atomic accumulators (gsum + acc2 are contiguous).
  hipMemsetAsync(gsum, 0, (16 + 16384) * sizeof(float), stream);

  k_fc1<<<1024, 256, 0, stream>>>(input, fc1_w, fc1_b, h);
  k_norm_h<<<2048, 256, 0, stream>>>(h);
  k_assign<<<2048, 64, 0, stream>>>(h, conv_w, conv_b, a);
  k_asum<<<16, 64, 0, stream>>>(a, asum);
  k_vlad<<<64, 256, 0, stream>>>(a, h, asum, cent, vlad);
  k_vlad_norm<<<128, 256, 0, stream>>>(vlad, gsum);
  k_fc2<<<128, 256, 0, stream>>>(vlad, fc2_w, acc2);
  k_final<<<64, 256, 0, stream>>>(acc2, gsum, fc2_b, out);
}